// PointNetSetAbstraction_9354438770914
// MI455X (gfx1250) — compile-verified
//
#include <hip/hip_runtime.h>
#include <hip/hip_bf16.h>

typedef __attribute__((ext_vector_type(2))) float v2f;
typedef __attribute__((ext_vector_type(4))) float v4f;
typedef __attribute__((ext_vector_type(8))) float v8f;
typedef int gccv4i __attribute__((vector_size(16)));   // matches builtin's param type

#define B_     8
#define N_     8192
#define S_     2048
#define K_     32
#define MTOT   (8L * 2048L * 32L)   // 524288 rows through the MLP
#define CONVBLOCKS 4096             // MTOT / 128

// ---------------------------------------------------------------------------
// CDNA5 async global->LDS copies (ASYNCcnt path). Probe-confirmed present on
// this toolchain; param0 = global src (addrspace(1), int-typed), param1 = LDS
// dst (addrspace(3)), then imm offset, imm cpol.
// ---------------------------------------------------------------------------
#if defined(__has_builtin)
#  if __has_builtin(__builtin_amdgcn_global_load_async_to_lds_b32)
#    define HAVE_ASYNC_B32 1
#  endif
#  if __has_builtin(__builtin_amdgcn_global_load_async_to_lds_b128)
#    define HAVE_ASYNC_B128 1
#  endif
#  if __has_builtin(__builtin_amdgcn_s_wait_asynccnt)
#    define HAVE_WAIT_ASYNC 1
#  endif
#endif

__device__ __forceinline__ void ld_async_f32(float* l, const float* g) {
#if defined(HAVE_ASYNC_B32)
    __builtin_amdgcn_global_load_async_to_lds_b32(
        (__attribute__((address_space(1))) int*)g,
        (__attribute__((address_space(3))) int*)l, 0, 0);
#else
    *l = *g;
#endif
}

__device__ __forceinline__ void ld_async_f32x4(float* l, const float* g) {
#if defined(HAVE_ASYNC_B128)
    __builtin_amdgcn_global_load_async_to_lds_b128(
        (__attribute__((address_space(1))) gccv4i*)g,
        (__attribute__((address_space(3))) gccv4i*)l, 0, 0);
#elif defined(HAVE_ASYNC_B32)
    ld_async_f32(l + 0, g + 0);
    ld_async_f32(l + 1, g + 1);
    ld_async_f32(l + 2, g + 2);
    ld_async_f32(l + 3, g + 3);
#else
    *(v4f*)l = *(const v4f*)g;
#endif
}

__device__ __forceinline__ void async_wait_all() {
#if defined(HAVE_ASYNC_B32) || defined(HAVE_ASYNC_B128)
#  if defined(HAVE_WAIT_ASYNC)
    __builtin_amdgcn_s_wait_asynccnt(0);
#  else
    asm volatile("s_wait_asynccnt 0" ::: "memory");
#  endif
#endif
}

// ---------------------------------------------------------------------------
// 1) Furthest point sampling: one block per batch. The 2048 rounds are an
//    inherent serial latency chain, so minimize per-round cost: points AND
//    running min-distances live in registers (8 per thread), argmax uses
//    shfl-down wave32 reduction; only TWO barriers per round. Broadcast slot
//    is parity double-buffered (wv/wi and bcast writes are separated from
//    their readers by the two barriers -> race-free, deterministic).
//    Tie-break: lower index wins (matches jnp.argmax first occurrence).
// ---------------------------------------------------------------------------
__global__ void fps_kernel(const float* __restrict__ xyz,   // [B,3,N]
                           float* __restrict__ new_xyz_t,   // [B,S,3] (ws)
                           float* __restrict__ out_xyz,     // [B,3,S] (d_out)
                           float* __restrict__ out_idx)     // [B,S] as float
{
    __shared__ float wv[32];
    __shared__ int   wi[32];
    __shared__ int   bcast[2];

    const int b    = blockIdx.x;
    const int t    = threadIdx.x;          // 1024 threads = 32 waves
    const int lane = t & 31;
    const int wave = t >> 5;
    const float* xb = xyz + (long)b * 3 * N_;

    float lx[8], ly[8], lz[8], ld[8];
#pragma unroll
    for (int j = 0; j < 8; ++j) {
        const int n = t + j * 1024;
        lx[j] = xb[n];
        ly[j] = xb[N_ + n];
        lz[j] = xb[2 * N_ + n];
        ld[j] = 1e10f;
    }

    int last = 0;
    for (int it = 0; it < S_; ++it) {
        const float cx = xb[last];
        const float cy = xb[N_ + last];
        const float cz = xb[2 * N_ + last];
        if (t == 0) {
            out_idx[(long)b * S_ + it] = (float)last;
            const long m = (long)b * S_ + it;
            new_xyz_t[m * 3 + 0] = cx;
            new_xyz_t[m * 3 + 1] = cy;
            new_xyz_t[m * 3 + 2] = cz;
            out_xyz[(long)b * 3 * S_ + 0 * S_ + it] = cx;
            out_xyz[(long)b * 3 * S_ + 1 * S_ + it] = cy;
            out_xyz[(long)b * 3 * S_ + 2 * S_ + it] = cz;
        }

        float bv = -1.0f; int bi = N_;
#pragma unroll
        for (int j = 0; j < 8; ++j) {
            const float dx = lx[j] - cx, dy = ly[j] - cy, dz = lz[j] - cz;
            const float d  = dx * dx + dy * dy + dz * dz;
            ld[j] = ld[j] < d ? ld[j] : d;
            if (ld[j] > bv) { bv = ld[j]; bi = t + j * 1024; } // strict > keeps low n
        }
        // wave32 argmax reduce (no barrier)
#pragma unroll
        for (int off = 16; off > 0; off >>= 1) {
            const float ov = __shfl_down(bv, off, 32);
            const int   oi = __shfl_down(bi, off, 32);
            if (ov > bv || (ov == bv && oi < bi)) { bv = ov; bi = oi; }
        }
        if (lane == 0) { wv[wave] = bv; wi[wave] = bi; }
        __syncthreads();
        if (wave == 0) {
            bv = wv[lane]; bi = wi[lane];
#pragma unroll
            for (int off = 16; off > 0; off >>= 1) {
                const float ov = __shfl_down(bv, off, 32);
                const int   oi = __shfl_down(bi, off, 32);
                if (ov > bv || (ov == bv && oi < bi)) { bv = ov; bi = oi; }
            }
            if (lane == 0) bcast[it & 1] = bi;
        }
        __syncthreads();
        last = bcast[it & 1];
    }
}

// ---------------------------------------------------------------------------
// 2) KNN (top-32 smallest of 8192 sq-dists) + gather + feature build.
//    Candidate points async-staged into LDS in two 4096-point tiles (48KB).
//    Per-thread replace-the-worst list; strict < keeps earlier index.
//    The neighbor SET is all that matters (max-pool over K and BN stats are
//    permutation invariant), so exact top_k ordering is not needed.
// ---------------------------------------------------------------------------
__global__ void knn_build_kernel(const float* __restrict__ xyz,       // [B,3,N]
                                 const float* __restrict__ points,    // [B,3,N]
                                 const float* __restrict__ new_xyz_t, // [B,S,3]
                                 float* __restrict__ x0)              // [MTOT,6]
{
    __shared__ __align__(16) float sx[4096], sy[4096], sz[4096];

    const int b  = blockIdx.y;
    const int s  = blockIdx.x * 256 + threadIdx.x;
    const long q = (long)b * S_ + s;
    const float qx = new_xyz_t[q * 3 + 0];
    const float qy = new_xyz_t[q * 3 + 1];
    const float qz = new_xyz_t[q * 3 + 2];
    const float* xb = xyz + (long)b * 3 * N_;
    const float* pb = points + (long)b * 3 * N_;

    float bd[K_]; int bi[K_];
    for (int j = 0; j < K_; ++j) { bd[j] = 1e30f; bi[j] = 0; }
    float wv = 1e30f; int ws = 0;

    for (int tile = 0; tile < 2; ++tile) {
        const int base = tile * 4096;
        __syncthreads();
        for (int i = threadIdx.x * 4; i < 4096; i += 256 * 4) {
            ld_async_f32x4(&sx[i], &xb[base + i]);
            ld_async_f32x4(&sy[i], &xb[N_ + base + i]);
            ld_async_f32x4(&sz[i], &xb[2 * N_ + base + i]);
        }
        async_wait_all();
        __syncthreads();
        for (int i = 0; i < 4096; ++i) {
            const float dx = sx[i] - qx, dy = sy[i] - qy, dz = sz[i] - qz;
            const float d = dx * dx + dy * dy + dz * dz;
            if (d < wv) {
                bd[ws] = d; bi[ws] = base + i;
                wv = bd[0]; ws = 0;
                for (int j = 1; j < K_; ++j)
                    if (bd[j] > wv) { wv = bd[j]; ws = j; }
            }
        }
    }

    for (int j = 0; j < K_; ++j) {
        const int id = bi[j];
        const long m = (q * K_ + j);
        x0[m * 6 + 0] = xb[id] - qx;
        x0[m * 6 + 1] = xb[N_ + id] - qy;
        x0[m * 6 + 2] = xb[2 * N_ + id] - qz;
        x0[m * 6 + 3] = pb[id];
        x0[m * 6 + 4] = pb[N_ + id];
        x0[m * 6 + 5] = pb[2 * N_ + id];
    }
}

// ---------------------------------------------------------------------------
// 3) 1x1 conv as GEMM via V_WMMA_F32_16X16X4_F32 (full fp32 precision),
//    with FUSED per-channel BN statistics (sum/sumsq of the raw outputs,
//    reduced per block in fixed order -> deterministic, no float atomics).
//    X tile is async-staged raw; BN(scale,shift)+ReLU of the PREVIOUS layer
//    is applied at A-fragment read time (params staged in LDS).
//    A (16x4 f32): lane<16 -> K={kb,kb+1}; lane>=16 -> K={kb+2,kb+3}.
//    B (4x16 f32): N = lane&15, same K half-split. C/D: M=r+8*(lane>=16).
// ---------------------------------------------------------------------------
template <int CIN, int CINP, int COUT, bool APPLY_BN>
__global__ void conv_wmma_kernel(const float* __restrict__ X,       // [MTOT,CIN]
                                 const float* __restrict__ W,       // [COUT,CIN]
                                 const float* __restrict__ bnscale, // [CIN]
                                 const float* __restrict__ bnshift, // [CIN]
                                 float* __restrict__ Y,             // [MTOT,COUT]
                                 float* __restrict__ partial)       // [CONVBLOCKS,COUT,2]
{
    __shared__ __align__(16) float Xs[128 * CINP];
    __shared__ float Ws[COUT * CINP];
    __shared__ float bns[CINP], bnh[CINP];
    __shared__ float wpart[8 * COUT * 2];   // per-wave per-channel (sum, sumsq)

    const int  tid   = threadIdx.x;
    const long mbase = (long)blockIdx.x * 128;

    for (int i = tid; i < COUT * CINP; i += 256) {
        const int o = i / CINP, k = i % CINP;
        Ws[i] = (k < CIN) ? W[o * CIN + k] : 0.0f;     // zero-pad K (6 -> 8)
    }
    if constexpr (APPLY_BN) {
        for (int i = tid; i < CIN; i += 256) { bns[i] = bnscale[i]; bnh[i] = bnshift[i]; }
    }
    if constexpr (CIN == CINP) {
        // contiguous tile: 128*CIN floats, vectorized async b128
        for (int i = tid * 4; i < 128 * CINP; i += 256 * 4)
            ld_async_f32x4(&Xs[i], &X[mbase * CIN + i]);
    } else {
        for (int i = tid; i < 128 * CINP; i += 256) {
            const int r = i / CINP, k = i % CINP;
            if (k < CIN) ld_async_f32(&Xs[i], &X[(mbase + r) * CIN + k]);
            else         Xs[i] = 0.0f;
        }
    }
    async_wait_all();
    __syncthreads();

    const int wave     = tid >> 5;
    const int lane     = tid & 31;
    const int laneHalf = lane >> 4;   // 0: K pair {0,1}; 1: K pair {2,3}
    const int lane15   = lane & 15;
    const int rowA     = wave * 16 + lane15;

    for (int nt = 0; nt < COUT / 16; ++nt) {
        v8f acc = {};
        for (int kc = 0; kc < CINP / 4; ++kc) {
            const int kb = kc * 4 + laneHalf * 2;
            v2f a, bm;
            float ax = Xs[rowA * CINP + kb];
            float ay = Xs[rowA * CINP + kb + 1];
            if constexpr (APPLY_BN) {
                ax = ax * bns[kb] + bnh[kb];         ax = ax > 0.0f ? ax : 0.0f;
                ay = ay * bns[kb + 1] + bnh[kb + 1]; ay = ay > 0.0f ? ay : 0.0f;
            }
            a.x = ax; a.y = ay;
            const int n = nt * 16 + lane15;
            bm.x = Ws[n * CINP + kb];                // B[k][n] = W[n][k]
            bm.y = Ws[n * CINP + kb + 1];
            acc = __builtin_amdgcn_wmma_f32_16x16x4_f32(
                false, a, false, bm, (short)0, acc, false, false);
        }
        // store tile + per-lane stats over its 8 rows
        float ls = 0.0f, lq = 0.0f;
        for (int r = 0; r < 8; ++r) {
            const long m = mbase + wave * 16 + r + laneHalf * 8;
            const int  n = nt * 16 + lane15;
            const float v = acc[r];
            Y[m * COUT + n] = v;
            ls += v; lq += v * v;
        }
        // lanes L and L+16 hold the same channel -> combine halves
        ls += __shfl_down(ls, 16, 32);
        lq += __shfl_down(lq, 16, 32);
        if (lane < 16) {
            wpart[(wave * COUT + nt * 16 + lane15) * 2 + 0] = ls;
            wpart[(wave * COUT + nt * 16 + lane15) * 2 + 1] = lq;
        }
    }
    __syncthreads();
    if (tid < COUT) {
        float ts = 0.0f, tq = 0.0f;
        for (int w = 0; w < 8; ++w) {               // fixed order
            ts += wpart[(w * COUT + tid) * 2 + 0];
            tq += wpart[(w * COUT + tid) * 2 + 1];
        }
        partial[((long)blockIdx.x * COUT + tid) * 2 + 0] = ts;
        partial[((long)blockIdx.x * COUT + tid) * 2 + 1] = tq;
    }
}

// ---------------------------------------------------------------------------
// 4) BN stat finalize: fixed-order sum of per-block partials -> scale/shift.
// ---------------------------------------------------------------------------
template <int C>
__global__ void bn_finalize_kernel(const float* __restrict__ partial, int chunks,
                                   const float* __restrict__ gamma,
                                   const float* __restrict__ beta,
                                   float* __restrict__ scale,
                                   float* __restrict__ shift)
{
    const int c = threadIdx.x;
    if (c < C) {
        float s = 0.0f, q = 0.0f;
        for (int ch = 0; ch < chunks; ++ch) {       // fixed order
            s += partial[((long)ch * C + c) * 2 + 0];
            q += partial[((long)ch * C + c) * 2 + 1];
        }
        const float invM = 1.0f / (float)MTOT;
        const float mean = s * invM;
        const float var  = q * invM - mean * mean;
        const float istd = rsqrtf(var + 1e-5f);
        const float sc   = gamma[c] * istd;
        scale[c] = sc;
        shift[c] = beta[c] - mean * sc;
    }
}

// ---------------------------------------------------------------------------
// 5) BN3 + ReLU + max over K neighbors; write transposed [B,64,S] into d_out.
// ---------------------------------------------------------------------------
__global__ void final_max_kernel(const float* __restrict__ Y3,     // [MTOT,64]
                                 const float* __restrict__ scale,
                                 const float* __restrict__ shift,
                                 float* __restrict__ out_points)   // [B,64,S]
{
    const int  c  = threadIdx.x;       // 64
    const long bs = blockIdx.x;        // B*S
    const float sc = scale[c], sh = shift[c];
    float best = -1e30f;
    for (int k = 0; k < K_; ++k) {
        float v = Y3[(bs * K_ + k) * 64 + c];
        v = v * sc + sh;
        v = v > 0.0f ? v : 0.0f;
        best = best > v ? best : v;
    }
    const long b = bs / S_, s = bs % S_;
    out_points[(b * 64 + c) * S_ + s] = best;
}

// ---------------------------------------------------------------------------
extern "C" void kernel_launch(void* const* d_in, const int* in_sizes, int n_in,
                              void* d_out, int out_size, void* d_ws, size_t ws_size,
                              hipStream_t stream) {
    (void)in_sizes; (void)n_in; (void)out_size; (void)ws_size;

    const float* xyz    = (const float*)d_in[0];
    const float* points = (const float*)d_in[1];
    const float* w0 = (const float*)d_in[2];
    const float* g0 = (const float*)d_in[3];
    const float* b0 = (const float*)d_in[4];
    const float* w1 = (const float*)d_in[5];
    const float* g1 = (const float*)d_in[6];
    const float* b1 = (const float*)d_in[7];
    const float* w2 = (const float*)d_in[8];
    const float* g2 = (const float*)d_in[9];
    const float* b2 = (const float*)d_in[10];

    // workspace layout (floats)
    float* ws        = (float*)d_ws;
    float* new_xyz_t = ws;                               // B*S*3
    float* x0        = new_xyz_t + (long)B_ * S_ * 3;    // MTOT*6
    float* y1        = x0 + MTOT * 6;                    // MTOT*32
    float* y2        = y1 + MTOT * 32;                   // MTOT*32
    float* y3        = y2 + MTOT * 32;                   // MTOT*64
    float* partial   = y3 + MTOT * 64;                   // CONVBLOCKS*64*2
    float* bnp       = partial + (long)CONVBLOCKS * 64 * 2;
    float* sc0 = bnp,        * sh0 = bnp + 64;
    float* sc1 = bnp + 128,  * sh1 = bnp + 192;
    float* sc2 = bnp + 256,  * sh2 = bnp + 320;

    // d_out layout: new_xyz [B,3,S] | new_points [B,64,S] | fps_idx [B,S]
    float* out_xyz    = (float*)d_out;
    float* out_points = out_xyz + (long)B_ * 3 * S_;
    float* out_idx    = out_points + (long)B_ * 64 * S_;

    fps_kernel<<<B_, 1024, 0, stream>>>(xyz, new_xyz_t, out_xyz, out_idx);

    knn_build_kernel<<<dim3(S_ / 256, B_), 256, 0, stream>>>(xyz, points, new_xyz_t, x0);

    conv_wmma_kernel<6, 8, 32, false><<<CONVBLOCKS, 256, 0, stream>>>(x0, w0, nullptr, nullptr, y1, partial);
    bn_finalize_kernel<32><<<1, 32, 0, stream>>>(partial, CONVBLOCKS, g0, b0, sc0, sh0);

    conv_wmma_kernel<32, 32, 32, true><<<CONVBLOCKS, 256, 0, stream>>>(y1, w1, sc0, sh0, y2, partial);
    bn_finalize_kernel<32><<<1, 32, 0, stream>>>(partial, CONVBLOCKS, g1, b1, sc1, sh1);

    conv_wmma_kernel<32, 32, 64, true><<<CONVBLOCKS, 256, 0, stream>>>(y2, w2, sc1, sh1, y3, partial);
    bn_finalize_kernel<64><<<1, 64, 0, stream>>>(partial, CONVBLOCKS, g2, b2, sc2, sh2);

    final_max_kernel<<<(int)((long)B_ * S_), 64, 0, stream>>>(y3, sc2, sh2, out_points);
}